// DeformConv_9612136809139
// MI455X (gfx1250) — compile-verified
//
#include <hip/hip_runtime.h>

typedef float v2f __attribute__((ext_vector_type(2)));
typedef float v8f __attribute__((ext_vector_type(8)));

#define Hc   128
#define Wc   128
#define Cc   64
#define Oc   64
#define Kc   9          // 3x3 taps
#define KD   576        // C*K reduction depth
#define KCH  16         // K-chunk staged in LDS
#define NCH  36         // KD / KCH
#define PIXB 128        // pixels per block (one output row)
#define CSTR 136        // padded cols row stride (floats)

__global__ __launch_bounds__(256) void dcn_fused_wmma(
    const float* __restrict__ x, const float* __restrict__ off,
    const float* __restrict__ msk, const float* __restrict__ wgt,
    const float* __restrict__ bias, float* __restrict__ out)
{
    __shared__ float s_w4[Kc * PIXB * 4];    // bilinear weights (mask*valid folded)
    __shared__ int   s_a4[Kc * PIXB * 4];    // clamped plane element indices
    __shared__ float s_cols[KCH][CSTR];      // cols tile: [k-chunk][pixel]

    const int tid = threadIdx.x;
    const int blk = blockIdx.x;
    const int n   = blk >> 7;       // blk / 128
    const int pb  = blk & 127;
    const int p0  = pb * PIXB;

    const float* xn = x + (size_t)n * Cc * Hc * Wc;

    // ---- Phase 0: per-(tap,pixel) sampling metadata -> LDS ----
    for (int e = tid; e < Kc * PIXB; e += 256) {
        const int k  = e >> 7;          // tap 0..8
        const int p  = e & 127;
        const int pix = p0 + p;
        const int ho = pix >> 7;
        const int wo = pix & 127;

        const int ob = ((n * 2 * Kc + 2 * k) * Hc + ho) * Wc + wo;
        const float dy = off[ob];
        const float dx = off[ob + Hc * Wc];
        const float m  = msk[((n * Kc + k) * Hc + ho) * Wc + wo];

        const float ys = (float)(ho - 1 + k / 3) + dy;
        const float xs = (float)(wo - 1 + k % 3) + dx;
        const float y0f = floorf(ys), x0f = floorf(xs);
        const int iy0 = (int)y0f, ix0 = (int)x0f;
        const int iy1 = iy0 + 1,  ix1 = ix0 + 1;
        const float wy = ys - y0f, wx = xs - x0f;

        const bool vy0 = (iy0 >= 0) & (iy0 < Hc);
        const bool vy1 = (iy1 >= 0) & (iy1 < Hc);
        const bool vx0 = (ix0 >= 0) & (ix0 < Wc);
        const bool vx1 = (ix1 >= 0) & (ix1 < Wc);
        const int cy0 = min(max(iy0, 0), Hc - 1);
        const int cy1 = min(max(iy1, 0), Hc - 1);
        const int cx0 = min(max(ix0, 0), Wc - 1);
        const int cx1 = min(max(ix1, 0), Wc - 1);

        const int b4 = e * 4;
        s_w4[b4 + 0] = (vy0 & vx0) ? (1.f - wy) * (1.f - wx) * m : 0.f;
        s_w4[b4 + 1] = (vy0 & vx1) ? (1.f - wy) * wx * m : 0.f;
        s_w4[b4 + 2] = (vy1 & vx0) ? wy * (1.f - wx) * m : 0.f;
        s_w4[b4 + 3] = (vy1 & vx1) ? wy * wx * m : 0.f;
        s_a4[b4 + 0] = cy0 * Wc + cx0;
        s_a4[b4 + 1] = cy0 * Wc + cx1;
        s_a4[b4 + 2] = cy1 * Wc + cx0;
        s_a4[b4 + 3] = cy1 * Wc + cx1;
    }
    __syncthreads();

    // ---- WMMA lane decomposition (wave32) ----
    const int wave  = tid >> 5;     // 8 waves -> 8 pixel tiles of 16
    const int lane  = tid & 31;
    const int laneM = lane & 15;
    const int hi    = lane >> 4;    // selects K-pair / M-upper-half per ISA layout
    const int pt    = wave;

    v8f acc[4];
#pragma unroll
    for (int ot = 0; ot < 4; ++ot) {
        v8f z = {0.f, 0.f, 0.f, 0.f, 0.f, 0.f, 0.f, 0.f};
        acc[ot] = z;
    }

    // ---- Phase 1: K-chunk loop: build cols tile, then WMMA ----
    for (int ch = 0; ch < NCH; ++ch) {
        for (int e = tid; e < KCH * PIXB; e += 256) {   // exactly 8 iters
            const int kk = e >> 7;
            const int p  = e & 127;
            const int cidx = ch * KCH + kk;
            const int c = cidx / 9;
            const int k = cidx - c * 9;
            const float* pl = xn + c * (Hc * Wc);
            const int b4 = (k * PIXB + p) * 4;
            const float v = s_w4[b4 + 0] * pl[s_a4[b4 + 0]]
                          + s_w4[b4 + 1] * pl[s_a4[b4 + 1]]
                          + s_w4[b4 + 2] * pl[s_a4[b4 + 2]]
                          + s_w4[b4 + 3] * pl[s_a4[b4 + 3]];
            s_cols[kk][p] = v;
        }
        __syncthreads();

        const int kb = ch * KCH;
#pragma unroll
        for (int ks = 0; ks < 4; ++ks) {
            const int kloc = ks * 4 + hi * 2;
            // B fragment: 4x16 tile, VGPR0 = rows {K0,K2}, VGPR1 = rows {K1,K3}
            v2f b;
            b.x = s_cols[kloc + 0][pt * 16 + laneM];
            b.y = s_cols[kloc + 1][pt * 16 + laneM];
#pragma unroll
            for (int ot = 0; ot < 4; ++ot) {
                // A fragment: 16x4 tile of weights, row M=laneM, K pair by `hi`
                const float* ap = wgt + (size_t)(ot * 16 + laneM) * KD + (kb + kloc);
                v2f a;
                a.x = ap[0];
                a.y = ap[1];
                acc[ot] = __builtin_amdgcn_wmma_f32_16x16x4_f32(
                    false, a, false, b, (short)0, acc[ot], false, false);
            }
        }
        __syncthreads();
    }

    // ---- Phase 2: write out (C/D layout: VGPR r -> M = r + hi*8) ----
    const int pix = p0 + pt * 16 + laneM;
#pragma unroll
    for (int ot = 0; ot < 4; ++ot) {
#pragma unroll
        for (int r = 0; r < 8; ++r) {
            const int o = ot * 16 + hi * 8 + r;
            out[((size_t)n * Oc + o) * (Hc * Wc) + pix] = acc[ot][r] + bias[o];
        }
    }
}

extern "C" void kernel_launch(void* const* d_in, const int* in_sizes, int n_in,
                              void* d_out, int out_size, void* d_ws, size_t ws_size,
                              hipStream_t stream) {
    const float* x    = (const float*)d_in[0];
    const float* off  = (const float*)d_in[1];
    const float* msk  = (const float*)d_in[2];
    const float* wgt  = (const float*)d_in[3];
    const float* bias = (const float*)d_in[4];
    float* out = (float*)d_out;
    (void)in_sizes; (void)n_in; (void)out_size; (void)d_ws; (void)ws_size;

    // 8 images * 128 row-tiles each = 1024 blocks of 256 threads (8 wave32)
    dim3 grid(8 * 128), block(256);
    hipLaunchKernelGGL(dcn_fused_wmma, grid, block, 0, stream,
                       x, off, msk, wgt, bias, out);
}